// Net_63806034149753
// MI455X (gfx1250) — compile-verified
//
#include <hip/hip_runtime.h>
#include <hip/hip_bf16.h>
#include <math.h>

typedef __attribute__((ext_vector_type(2))) float v2f;
typedef __attribute__((ext_vector_type(8))) float v8f;

#define N_FEAT_IN 512
#define N_HID 16
#define N_CLS 40

// ---------------------------------------------------------------------------
// Kernel 1: h = relu(x @ W1^T + b1) using V_WMMA_F32_16X16X4_F32.
// One wave32 computes one 16x16 output tile (16 rows x all 16 hidden feats).
// (Unchanged from round 1 — lowers to 4x v_wmma_f32_16x16x4_f32 clauses.)
// ---------------------------------------------------------------------------
__global__ void __launch_bounds__(256)
gemm1_relu_kernel(const float* __restrict__ x, const float* __restrict__ W1,
                  const float* __restrict__ b1, float* __restrict__ h, int n) {
    __shared__ float sW[N_HID * N_FEAT_IN]; // 32 KB of 320 KB LDS
    for (int i = threadIdx.x; i < N_HID * N_FEAT_IN; i += blockDim.x)
        sW[i] = W1[i];
    __syncthreads();

    const int wave = threadIdx.x >> 5;
    const int lane = threadIdx.x & 31;
    const int tile = blockIdx.x * 8 + wave;
    const int row0 = tile * 16;
    if (row0 >= n) return;

    const int half = lane >> 4;   // 0: K+0/1, 1: K+2/3
    const int l    = lane & 15;   // M index for A, N index for B

    const float* __restrict__ xrow = x + (size_t)(row0 + l) * N_FEAT_IN;
    const float* __restrict__ wrow = sW + (size_t)l * N_FEAT_IN;

    v8f c = {};
#pragma unroll 4
    for (int k = 0; k < N_FEAT_IN; k += 4) {
        const int ka = k + half * 2;
        v2f a, b;
        a[0] = xrow[ka];     a[1] = xrow[ka + 1];
        b[0] = wrow[ka];     b[1] = wrow[ka + 1];
        c = __builtin_amdgcn_wmma_f32_16x16x4_f32(
                /*neg_a=*/false, a, /*neg_b=*/false, b,
                /*c_mod=*/(short)0, c, /*reuse_a=*/false, /*reuse_b=*/false);
    }

    const float bias = b1[l];
#pragma unroll
    for (int j = 0; j < 8; ++j) {
        const int m = row0 + j + half * 8;
        float v = c[j] + bias;
        h[(size_t)m * N_HID + l] = v > 0.f ? v : 0.f;
    }
}

// ---------------------------------------------------------------------------
// Normalize rows: xn[i] = h[i]/r[i], r[i] = max(||h[i]||, eps); h == xn*r exact.
// ---------------------------------------------------------------------------
__global__ void normalize_kernel(const float* __restrict__ h,
                                 float* __restrict__ xn,
                                 float* __restrict__ r, int n) {
    int i = blockIdx.x * blockDim.x + threadIdx.x;
    if (i >= n) return;
    const float4* __restrict__ src = (const float4*)(h + (size_t)i * N_HID);
    float4* __restrict__ dstp = (float4*)(xn + (size_t)i * N_HID);
    float4 v[4];
    float ss = 0.f;
#pragma unroll
    for (int q = 0; q < 4; ++q) {
        v[q] = src[q];
        ss += v[q].x * v[q].x + v[q].y * v[q].y + v[q].z * v[q].z + v[q].w * v[q].w;
    }
    float rr = fmaxf(sqrtf(ss), 1e-12f);
    float inv = 1.f / rr;
    r[i] = rr;
#pragma unroll
    for (int q = 0; q < 4; ++q) {
        float4 o = v[q];
        o.x *= inv; o.y *= inv; o.z *= inv; o.w *= inv;
        dstp[q] = o;
    }
}

// ---------------------------------------------------------------------------
// CSR build: degree (self-loop folded in as init=1), 3-kernel scan, scatter.
// ---------------------------------------------------------------------------
__global__ void init_deg_kernel(int* __restrict__ deg, int n) {
    int i = blockIdx.x * blockDim.x + threadIdx.x;
    if (i < n) deg[i] = 1;  // self-loop
}

__global__ void count_deg_kernel(const long long* __restrict__ dst,
                                 int* __restrict__ deg, long long E) {
    long long e = (long long)blockIdx.x * blockDim.x + threadIdx.x;
    if (e < E) atomicAdd(deg + (int)dst[e], 1);
}

__global__ void __launch_bounds__(256)
scan_block_kernel(const int* __restrict__ deg, int* __restrict__ part,
                  int* __restrict__ blkSum, int n) {
    __shared__ int tmp[256];
    int i = blockIdx.x * 256 + threadIdx.x;
    int v = (i < n) ? deg[i] : 0;
    tmp[threadIdx.x] = v;
    __syncthreads();
    for (int off = 1; off < 256; off <<= 1) {
        int t = (threadIdx.x >= off) ? tmp[threadIdx.x - off] : 0;
        __syncthreads();
        tmp[threadIdx.x] += t;
        __syncthreads();
    }
    if (i < n) part[i] = tmp[threadIdx.x] - v;          // exclusive
    if (threadIdx.x == 255) blkSum[blockIdx.x] = tmp[255];
}

__global__ void __launch_bounds__(1024)
scan_tops_kernel(const int* __restrict__ blkSum, int* __restrict__ blkOff, int nb) {
    __shared__ int tmp[1024];
    int v = ((int)threadIdx.x < nb) ? blkSum[threadIdx.x] : 0;
    tmp[threadIdx.x] = v;
    __syncthreads();
    for (int off = 1; off < 1024; off <<= 1) {
        int t = (threadIdx.x >= (unsigned)off) ? tmp[threadIdx.x - off] : 0;
        __syncthreads();
        tmp[threadIdx.x] += t;
        __syncthreads();
    }
    if ((int)threadIdx.x < nb) blkOff[threadIdx.x] = tmp[threadIdx.x] - v; // exclusive
}

__global__ void scan_add_kernel(const int* __restrict__ part,
                                const int* __restrict__ blkOff,
                                int* __restrict__ offs, int* __restrict__ cursor,
                                int n, int etot) {
    int i = blockIdx.x * blockDim.x + threadIdx.x;
    if (i < n) {
        int o = part[i] + blkOff[i >> 8];
        offs[i] = o;
        cursor[i] = o;
    }
    if (i == 0) offs[n] = etot;
}

__global__ void scatter_kernel(const long long* __restrict__ src,
                               const long long* __restrict__ dst,
                               int* __restrict__ cursor, int* __restrict__ csr,
                               long long E, int n) {
    long long e = (long long)blockIdx.x * blockDim.x + threadIdx.x;
    long long Etot = E + (long long)n;
    if (e >= Etot) return;
    int s, d;
    if (e < E) { s = (int)src[e]; d = (int)dst[e]; }
    else       { s = d = (int)(e - E); }
    int p = atomicAdd(cursor + d, 1);
    csr[p] = s;
}

// ---------------------------------------------------------------------------
// AGNN layer: one thread per destination node; online softmax over in-edges.
// Reads xn[src] (64B, L2-resident) + r[src]; zero float atomics.
// Outputs next layer's (xn_out, r_out) with h_out == xn_out * r_out exactly.
// ---------------------------------------------------------------------------
__global__ void __launch_bounds__(256)
agnn_layer_kernel(const int* __restrict__ offs, const int* __restrict__ csr,
                  const float* __restrict__ xn, const float* __restrict__ r,
                  float* __restrict__ xn_out, float* __restrict__ r_out,
                  int n, const float* __restrict__ beta_ptr) {
    int d = blockIdx.x * blockDim.x + threadIdx.x;
    if (d >= n) return;
    const float beta = beta_ptr ? beta_ptr[0] : 1.0f;

    float xd[N_HID];
    {
        const float4* __restrict__ p = (const float4*)(xn + (size_t)d * N_HID);
#pragma unroll
        for (int q = 0; q < 4; ++q) {
            float4 t = p[q];
            xd[q * 4 + 0] = t.x; xd[q * 4 + 1] = t.y;
            xd[q * 4 + 2] = t.z; xd[q * 4 + 3] = t.w;
        }
    }

    const int beg = offs[d], end = offs[d + 1];
    float m = -INFINITY, ssum = 0.f;
    float acc[N_HID];
#pragma unroll
    for (int j = 0; j < N_HID; ++j) acc[j] = 0.f;

    for (int p = beg; p < end; ++p) {
        int s = csr[p];
        if (p + 1 < end) {
            int s2 = csr[p + 1];
            __builtin_prefetch(xn + (size_t)s2 * N_HID, 0, 3); // global_prefetch_b8
        }
        float xs[N_HID];
        float dot = 0.f;
        {
            const float4* __restrict__ sp = (const float4*)(xn + (size_t)s * N_HID);
#pragma unroll
            for (int q = 0; q < 4; ++q) {
                float4 t = sp[q];
                xs[q * 4 + 0] = t.x; xs[q * 4 + 1] = t.y;
                xs[q * 4 + 2] = t.z; xs[q * 4 + 3] = t.w;
            }
#pragma unroll
            for (int j = 0; j < N_HID; ++j) dot += xs[j] * xd[j];
        }
        float a = beta * dot;
        if (a > m) {                        // rescale running state
            float c = __expf(m - a);        // exp(-inf)=0 handles first edge
            ssum *= c;
#pragma unroll
            for (int j = 0; j < N_HID; ++j) acc[j] *= c;
            m = a;
        }
        float w = __expf(a - m);
        ssum += w;
        float wr = w * r[s];                // h[src] == xn[src]*r[src]
#pragma unroll
        for (int j = 0; j < N_HID; ++j) acc[j] += wr * xs[j];
    }

    const float inv = 1.f / ssum;           // ssum >= exp(0) from self-loop
    float ss2 = 0.f;
#pragma unroll
    for (int j = 0; j < N_HID; ++j) {
        float f = acc[j] * inv;
        acc[j] = f;
        ss2 += f * f;
    }
    const float rr = fmaxf(sqrtf(ss2), 1e-12f);
    const float invn = 1.f / rr;
    r_out[d] = rr;
    float4* __restrict__ op = (float4*)(xn_out + (size_t)d * N_HID);
#pragma unroll
    for (int q = 0; q < 4; ++q) {
        float4 o;
        o.x = acc[q * 4 + 0] * invn; o.y = acc[q * 4 + 1] * invn;
        o.z = acc[q * 4 + 2] * invn; o.w = acc[q * 4 + 3] * invn;
        op[q] = o;
    }
}

// ---------------------------------------------------------------------------
// Head: logits = (xn*r) @ W2^T + b2 ; out = log_softmax(logits).
// ---------------------------------------------------------------------------
__global__ void __launch_bounds__(256)
head_kernel(const float* __restrict__ xn, const float* __restrict__ r,
            const float* __restrict__ W2, const float* __restrict__ b2,
            float* __restrict__ out, int n) {
    __shared__ float sW[N_CLS * N_HID];
    __shared__ float sb[N_CLS];
    for (int i = threadIdx.x; i < N_CLS * N_HID; i += blockDim.x) sW[i] = W2[i];
    for (int i = threadIdx.x; i < N_CLS; i += blockDim.x) sb[i] = b2[i];
    __syncthreads();

    int i = blockIdx.x * blockDim.x + threadIdx.x;
    if (i >= n) return;
    const float ri = r[i];
    float hv[N_HID];
    const float4* __restrict__ hp = (const float4*)(xn + (size_t)i * N_HID);
#pragma unroll
    for (int q = 0; q < 4; ++q) {
        float4 t = hp[q];
        hv[q * 4 + 0] = t.x * ri; hv[q * 4 + 1] = t.y * ri;
        hv[q * 4 + 2] = t.z * ri; hv[q * 4 + 3] = t.w * ri;
    }
    float logits[N_CLS];
    float mx = -INFINITY;
#pragma unroll 4
    for (int c = 0; c < N_CLS; ++c) {
        float acc = sb[c];
#pragma unroll
        for (int j = 0; j < N_HID; ++j) acc += hv[j] * sW[c * N_HID + j];
        logits[c] = acc;
        mx = fmaxf(mx, acc);
    }
    float se = 0.f;
#pragma unroll 4
    for (int c = 0; c < N_CLS; ++c) se += __expf(logits[c] - mx);
    const float lse = mx + __logf(se);
    float* __restrict__ op = out + (size_t)i * N_CLS;
#pragma unroll 4
    for (int c = 0; c < N_CLS; ++c) op[c] = logits[c] - lse;
}

// ---------------------------------------------------------------------------
// Launch
// ---------------------------------------------------------------------------
extern "C" void kernel_launch(void* const* d_in, const int* in_sizes, int n_in,
                              void* d_out, int out_size, void* d_ws, size_t ws_size,
                              hipStream_t stream) {
    const float*     x     = (const float*)d_in[0];
    const long long* ei    = (const long long*)d_in[1];
    const float*     W1    = (const float*)d_in[2];
    const float*     b1    = (const float*)d_in[3];
    const float*     W2    = (const float*)d_in[4];
    const float*     b2    = (const float*)d_in[5];
    const float*     beta2 = (const float*)d_in[6];
    float* out = (float*)d_out;

    const int n = in_sizes[0] / N_FEAT_IN;            // 100000
    const long long E = (long long)(in_sizes[1] / 2); // 6400000
    const long long Etot = E + (long long)n;
    const long long* srcp = ei;
    const long long* dstp = ei + E;

    // Workspace layout (4-byte units, 16B-aligned sections)
    float* ws = (float*)d_ws;
    size_t o = 0;
    float* hA     = ws + o; o += (size_t)n * N_HID;
    float* xnA    = ws + o; o += (size_t)n * N_HID;
    float* rA     = ws + o; o += (size_t)n;
    float* xnB    = ws + o; o += (size_t)n * N_HID;
    float* rB     = ws + o; o += (size_t)n;
    int*   deg    = (int*)(ws + o); o += (size_t)n;
    int*   part   = (int*)(ws + o); o += (size_t)n;
    int*   blkSum = (int*)(ws + o); o += 1024;
    int*   blkOff = (int*)(ws + o); o += 1024;
    int*   offs   = (int*)(ws + o); o += (size_t)n + 4;
    int*   cursor = (int*)(ws + o); o += (size_t)n;
    int*   csr    = (int*)(ws + o); o += (size_t)Etot;

    const int tiles      = (n + 15) / 16;
    const int gemmBlocks = (tiles + 7) / 8;
    const int nodeBlocks = (n + 255) / 256;
    const int nb         = (n + 255) / 256;           // scan blocks (391 <= 1024)
    const int edgeBlocks = (int)((E + 255) / 256);
    const int etotBlocks = (int)((Etot + 255) / 256);

    // 1) h = relu(x W1^T + b1)  [v_wmma_f32_16x16x4_f32]
    gemm1_relu_kernel<<<gemmBlocks, 256, 0, stream>>>(x, W1, b1, hA, n);
    normalize_kernel<<<nodeBlocks, 256, 0, stream>>>(hA, xnA, rA, n);

    // 2) CSR-by-dst build (once): degree -> scan -> scatter
    init_deg_kernel<<<nodeBlocks, 256, 0, stream>>>(deg, n);
    count_deg_kernel<<<edgeBlocks, 256, 0, stream>>>(dstp, deg, E);
    scan_block_kernel<<<nb, 256, 0, stream>>>(deg, part, blkSum, n);
    scan_tops_kernel<<<1, 1024, 0, stream>>>(blkSum, blkOff, nb);
    scan_add_kernel<<<nodeBlocks, 256, 0, stream>>>(part, blkOff, offs, cursor,
                                                    n, (int)Etot);
    scatter_kernel<<<etotBlocks, 256, 0, stream>>>(srcp, dstp, cursor, csr, E, n);

    // 3) AGNN layer 1 (beta = 1) and layer 2 (beta = beta2[0]), gather-only
    agnn_layer_kernel<<<nodeBlocks, 256, 0, stream>>>(offs, csr, xnA, rA,
                                                      xnB, rB, n, nullptr);
    agnn_layer_kernel<<<nodeBlocks, 256, 0, stream>>>(offs, csr, xnB, rB,
                                                      xnA, rA, n, beta2);

    // 4) logits + log_softmax
    head_kernel<<<nodeBlocks, 256, 0, stream>>>(xnA, rA, W2, b2, out, n);
}